// SparseLinearTorch_53515292508416
// MI455X (gfx1250) — compile-verified
//
#include <hip/hip_runtime.h>

#define OUT_F 4096
#define IN_F  4096
#define NNZ_TOT 1048576
#define BATCH 128
#define KC 64
#define LDS_STRIDE 65   // 64 + 1 pad -> conflict-free LDS banks

typedef __attribute__((ext_vector_type(2))) float v2f;
typedef __attribute__((ext_vector_type(8))) float v8f;

// ---------------- zero scratch (float4 stores) ----------------
__global__ void zero_f32x4(float4* __restrict__ p, int n4) {
    int i = blockIdx.x * blockDim.x + threadIdx.x;
    if (i < n4) p[i] = make_float4(0.f, 0.f, 0.f, 0.f);
}

// ---------------- COO -> dense scatter (handles dup coords like segment_sum) ----------------
__global__ void coo_densify(const float* __restrict__ vals,
                            const int* __restrict__ rows,
                            const int* __restrict__ cols,
                            float* __restrict__ Wd, int nnz) {
    int k = blockIdx.x * blockDim.x + threadIdx.x;
    if (k < nnz) {
        atomicAdd(&Wd[rows[k] * IN_F + cols[k]], vals[k]);
    }
}

// ---------------- dense fp32 WMMA GEMM: out[b,o] = sum_c X[b,c] * Wd[o,c] ----------------
// Grid: OUT_F/16 = 256 WGs. Block: 256 threads = 8 waves (wave32).
// Wave w computes the 16x16 tile (M-rows 16w..16w+15) x (N-cols n0..n0+15).
__global__ __launch_bounds__(256) void wmma_gemm_f32(const float* __restrict__ X,
                                                     const float* __restrict__ Wd,
                                                     float* __restrict__ out) {
    __shared__ float Xs[BATCH * LDS_STRIDE];  // 128 x KC tile of X  (33 KB)
    __shared__ float Bs[16 * LDS_STRIDE];     // 16  x KC tile of Wd ( 4 KB)

    const int n0    = blockIdx.x * 16;
    const int tid   = threadIdx.x;
    const int wave  = tid >> 5;        // 0..7 -> M tile index
    const int lane  = tid & 31;
    const int l16   = lane & 15;       // row-in-tile / col-in-tile
    const int khalf = (lane >> 4) * 2; // lanes 0-15 hold K={0,1}, lanes 16-31 K={2,3}
    const int mbase = wave * 16;

    v8f acc = {};

    for (int k0 = 0; k0 < IN_F; k0 += KC) {
        // Stage X block [128 x KC] (coalesced rows of 64 floats)
        for (int idx = tid; idx < BATCH * KC; idx += 256) {
            int m = idx >> 6;           // /KC
            int k = idx & (KC - 1);
            Xs[m * LDS_STRIDE + k] = X[m * IN_F + k0 + k];
        }
        // Stage Wd block [16 x KC] (coalesced); prefetch next K-chunk
        for (int idx = tid; idx < 16 * KC; idx += 256) {
            int n = idx >> 6;
            int k = idx & (KC - 1);
            const float* src = &Wd[(n0 + n) * IN_F + k0 + k];
            Bs[n * LDS_STRIDE + k] = *src;
            __builtin_prefetch(src + KC, 0, 0);   // global_prefetch_b8 for next chunk
        }
        __syncthreads();

        // 16 x V_WMMA_F32_16X16X4_F32 per chunk per wave
        #pragma unroll
        for (int kk = 0; kk < KC; kk += 4) {
            // A fragment (16x4, M x K): lane l -> M = l%16; VGPR0/1 -> K pair per half-wave
            const float* xa = &Xs[(mbase + l16) * LDS_STRIDE + kk + khalf];
            v2f a; a.x = xa[0]; a.y = xa[1];
            // B fragment (4x16, K x N): lane l -> N = l%16; same K-pair striping
            const float* bb = &Bs[l16 * LDS_STRIDE + kk + khalf];
            v2f b; b.x = bb[0]; b.y = bb[1];
            acc = __builtin_amdgcn_wmma_f32_16x16x4_f32(
                /*neg_a=*/false, a, /*neg_b=*/false, b,
                /*c_mod=*/(short)0, acc, /*reuse_a=*/false, /*reuse_b=*/false);
        }
        __syncthreads();
    }

    // C/D layout: lanes 0-15 -> M=r, lanes 16-31 -> M=8+r; N = lane%16
    const int mrow = mbase + (lane >> 4) * 8;
    const int ocol = n0 + l16;
    #pragma unroll
    for (int r = 0; r < 8; ++r) {
        out[(mrow + r) * OUT_F + ocol] = acc[r];
    }
}

// ---------------- fallback: direct atomic scatter (used only if ws too small) ----------------
__global__ void coo_atomic_fallback(const float* __restrict__ X,
                                    const float* __restrict__ vals,
                                    const int* __restrict__ rows,
                                    const int* __restrict__ cols,
                                    float* __restrict__ out) {
    long long t = (long long)blockIdx.x * blockDim.x + threadIdx.x;
    int k = (int)(t >> 5);
    int j = (int)(t & 31);
    if (k < NNZ_TOT) {
        float v = vals[k];
        int r = rows[k];
        int c = cols[k];
        for (int b = j; b < BATCH; b += 32) {
            atomicAdd(&out[b * OUT_F + r], v * X[b * IN_F + c]);
        }
    }
}

extern "C" void kernel_launch(void* const* d_in, const int* in_sizes, int n_in,
                              void* d_out, int out_size, void* d_ws, size_t ws_size,
                              hipStream_t stream) {
    (void)in_sizes; (void)n_in; (void)out_size;
    const float* X    = (const float*)d_in[0];
    const float* Wv   = (const float*)d_in[1];
    const int*   Wr   = (const int*)d_in[2];
    const int*   Wc   = (const int*)d_in[3];
    float*       out  = (float*)d_out;

    const size_t need = (size_t)OUT_F * IN_F * sizeof(float);  // 64 MB dense W

    if (ws_size >= need) {
        float* Wd = (float*)d_ws;

        // 1) zero dense W (16.7M floats -> 4.19M float4)
        int n4 = (OUT_F * IN_F) / 4;
        zero_f32x4<<<(n4 + 255) / 256, 256, 0, stream>>>((float4*)Wd, n4);

        // 2) scatter COO into dense W (atomic add handles duplicate coordinates)
        coo_densify<<<NNZ_TOT / 256, 256, 0, stream>>>(Wv, Wr, Wc, Wd, NNZ_TOT);

        // 3) dense fp32 WMMA GEMM; Wd stays L2-resident (64 MB << 192 MB L2)
        wmma_gemm_f32<<<OUT_F / 16, 256, 0, stream>>>(X, Wd, out);
    } else {
        // Fallback: zero out, then direct atomic scatter.
        int n4 = (BATCH * OUT_F) / 4;
        zero_f32x4<<<(n4 + 255) / 256, 256, 0, stream>>>((float4*)out, n4);

        long long threads = (long long)NNZ_TOT * 32;
        int blocks = (int)((threads + 255) / 256);
        coo_atomic_fallback<<<blocks, 256, 0, stream>>>(X, Wv, Wr, Wc, out);
    }
}